// Focal_Loss_58995670778177
// MI455X (gfx1250) — compile-verified
//
#include <hip/hip_runtime.h>
#include <cstdint>

#define DEV __device__ __forceinline__

typedef float v2f __attribute__((ext_vector_type(2)));
typedef float v8f __attribute__((ext_vector_type(8)));

constexpr float POS_T = 0.5f, NEG_T = 0.4f;
constexpr float LN2   = 0.69314718055994530942f;
constexpr int   C_CLS = 80;          // classes (fixed by reference)
constexpr int   TILE_A  = 128;       // anchors per tile
constexpr int   TILE_F4 = TILE_A * (C_CLS / 4);   // 2560 float4 = 40KB

// focal term for target==0 (and the bulk stream): 0.75 * p^2 * (-ln(1-p))
DEV float clampp(float x) { return fminf(fmaxf(x, 1e-4f), 1.0f - 1e-4f); }
DEV float fneg(float x) {
    float p = clampp(x);
    return (0.75f * LN2) * p * p * (-__log2f(1.0f - p));
}
// focal term for target==1: 0.25 * (1-p)^2 * (-ln p)
DEV float fpos(float x) {
    float p = clampp(x);
    float q = 1.0f - p;
    return (0.25f * LN2) * q * q * (-__log2f(p));
}

DEV float waveRed(float v) {
#pragma unroll
    for (int o = 16; o > 0; o >>= 1) v += __shfl_down(v, o, 32);
    return v;
}

// ---- CDNA5 async global->LDS copy (ASYNCcnt-tracked), per CDNA5 ISA ----
DEV void async_copy_b128(uint32_t lds_byte_addr, const void* gaddr) {
    asm volatile("global_load_async_to_lds_b128 %0, %1, off"
                 :: "v"(lds_byte_addr),
                    "v"((unsigned long long)(uintptr_t)gaddr)
                 : "memory");
}
DEV void wait_async0() {
    asm volatile("s_wait_asynccnt 0" ::: "memory");
}

// =====================================================================
// Kernel A: per-anchor assignment (IoU argmax), mask write, sparse
// positive-anchor corrections + regression smooth-L1. Deterministic
// block-level partials, no atomics.
// wsA[blk*3 + {0,1,2}] = {num_pos, cls_correction, reg_sum}
// =====================================================================
__global__ void __launch_bounds__(256)
assign_kernel(const float* __restrict__ cls,
              const float* __restrict__ reg,
              const float* __restrict__ anc,
              const float* __restrict__ lab,
              float* __restrict__ tmask,
              float* __restrict__ wsA,
              int A, int M, int nBlkA)
{
    __shared__ float sLab[64 * 5];
    __shared__ float sRed[8][3];

    const int b   = blockIdx.y;
    const int tid = threadIdx.x;
    const int a   = blockIdx.x * 256 + tid;
    const int nl  = M * 5;
    for (int i = tid; i < nl; i += 256) sLab[i] = lab[(size_t)b * nl + i];
    __syncthreads();

    float np = 0.f, corr = 0.f, rs = 0.f;
    if (a < A) {
        const float4 an = ((const float4*)anc)[a];
        const float aw = an.z - an.x, ah = an.w - an.y;
        const float acx = an.x + 0.5f * aw, acy = an.y + 0.5f * ah;
        const float areaA = aw * ah;

        float best = -1e30f; int bi = 0;
        for (int j = 0; j < M; ++j) {
            const float bx1 = sLab[j*5+0], by1 = sLab[j*5+1];
            const float bx2 = sLab[j*5+2], by2 = sLab[j*5+3];
            const float bc  = sLab[j*5+4];
            float iou;
            if (bc == -1.0f) {
                iou = -1.0f;                       // invalid boxes never win
            } else {
                float iw = fmaxf(fminf(an.z, bx2) - fmaxf(an.x, bx1), 0.f);
                float ih = fmaxf(fminf(an.w, by2) - fmaxf(an.y, by1), 0.f);
                float inter = iw * ih;
                float areaB = (bx2 - bx1) * (by2 - by1);
                float ua = fmaxf(areaA + areaB - inter, 1e-8f);
                iou = inter / ua;
            }
            if (iou > best) { best = iou; bi = j; }  // first-max = jnp.argmax
        }
        const bool pos = best >= POS_T;
        const bool ign = (!pos) && (best >= NEG_T);
        tmask[(size_t)b * A + a] = ign ? 0.0f : 1.0f;

        if (pos) {
            np = 1.0f;
            const int k = (int)sLab[bi*5+4];
            const float pk = cls[((size_t)b * A + a) * C_CLS + k];
            corr = fpos(pk) - fneg(pk);            // swap neg-term for pos-term at class k

            const float bx1 = sLab[bi*5+0], by1 = sLab[bi*5+1];
            const float bx2 = sLab[bi*5+2], by2 = sLab[bi*5+3];
            const float gwr = bx2 - bx1, ghr = by2 - by1;
            const float gcx = bx1 + 0.5f * gwr, gcy = by1 + 0.5f * ghr;
            const float gw = fmaxf(gwr, 1.0f), gh = fmaxf(ghr, 1.0f);
            const float t0 = ((gcx - acx) / aw) * 10.0f;   // /0.1
            const float t1 = ((gcy - acy) / ah) * 10.0f;   // /0.1
            const float t2 = __logf(gw / aw) * 5.0f;       // /0.2
            const float t3 = __logf(gh / ah) * 5.0f;       // /0.2
            const float4 rg = ((const float4*)reg)[(size_t)b * A + a];
            const float th9 = 1.0f / 9.0f, c9 = 0.5f / 9.0f;
            float d0 = fabsf(t0 - rg.x), d1 = fabsf(t1 - rg.y);
            float d2 = fabsf(t2 - rg.z), d3 = fabsf(t3 - rg.w);
            rs  = (d0 <= th9) ? 4.5f * d0 * d0 : d0 - c9;
            rs += (d1 <= th9) ? 4.5f * d1 * d1 : d1 - c9;
            rs += (d2 <= th9) ? 4.5f * d2 * d2 : d2 - c9;
            rs += (d3 <= th9) ? 4.5f * d3 * d3 : d3 - c9;
        }
    }
    np = waveRed(np); corr = waveRed(corr); rs = waveRed(rs);
    const int lane = tid & 31, wid = tid >> 5;
    if (lane == 0) { sRed[wid][0] = np; sRed[wid][1] = corr; sRed[wid][2] = rs; }
    __syncthreads();
    if (tid == 0) {
        float s0 = 0.f, s1 = 0.f, s2 = 0.f;
        for (int w = 0; w < 8; ++w) { s0 += sRed[w][0]; s1 += sRed[w][1]; s2 += sRed[w][2]; }
        const int blk = b * nBlkA + blockIdx.x;
        wsA[blk*3+0] = s0; wsA[blk*3+1] = s1; wsA[blk*3+2] = s2;
    }
}

// =====================================================================
// Kernel B: stream all 307MB of classifications through LDS with CDNA5
// async-to-LDS b128 copies; accumulate masked f_neg. Block reduction of
// the 256 per-thread partials done with V_WMMA_F32_16X16X4_F32:
// D = A(16x4) x ones(4x16) + C  ==> per-row sums; 4 chained WMMAs
// consume all 256 values (8 per lane of wave 0) at full f32 precision.
// =====================================================================
__global__ void __launch_bounds__(256)
focal_neg_kernel(const float* __restrict__ cls,
                 const float* __restrict__ tmask,
                 float* __restrict__ wsB,
                 int A, int numTiles)
{
    __shared__ float4 sTile[TILE_F4];    // 40KB tile: 128 anchors x 80 classes
    __shared__ float  sMask[TILE_A];
    __shared__ float  sAll[256];

    const int b = blockIdx.y, tile = blockIdx.x, tid = threadIdx.x;
    const int aBase = tile * TILE_A;
    const int nA  = min(TILE_A, A - aBase);
    const int nF4 = nA * (C_CLS / 4);

    const float4* gp = (const float4*)(cls + ((size_t)b * A + aBase) * C_CLS);
    const uint32_t ldsBase = (uint32_t)(uintptr_t)(&sTile[0]);

#pragma unroll
    for (int i = 0; i < TILE_F4 / 256; ++i) {        // 10 async b128 per lane
        const int idx4 = tid + i * 256;
        if (idx4 < nF4)
            async_copy_b128(ldsBase + (uint32_t)idx4 * 16u, (const void*)(gp + idx4));
    }
    if (tid < TILE_A)
        sMask[tid] = (tid < nA) ? tmask[(size_t)b * A + aBase + tid] : 0.0f;

    wait_async0();           // drain this wave's ASYNCcnt
    __syncthreads();         // all waves' tiles visible

    float acc = 0.f;
#pragma unroll
    for (int i = 0; i < TILE_F4 / 256; ++i) {
        const int idx4 = tid + i * 256;
        if (idx4 < nF4) {
            const float4 v = sTile[idx4];
            const float  m = sMask[idx4 / (C_CLS / 4)];
            const float  s = fneg(v.x) + fneg(v.y) + fneg(v.z) + fneg(v.w);
            acc += (m != 0.0f) ? s : 0.0f;   // select (not multiply): NaN-safe masking
        }
    }
    sAll[tid] = acc;
    __syncthreads();

    if (tid < 32) {                       // wave 0, EXEC all-ones (uniform branch)
        v8f d = {};                       // C/D accumulator (8 VGPRs)
        const v2f ones = {1.0f, 1.0f};    // B = ones(4x16)
#pragma unroll
        for (int t = 0; t < 4; ++t) {
            v2f a;
            a.x = sAll[tid * 8 + 2 * t + 0];
            a.y = sAll[tid * 8 + 2 * t + 1];
            // emits v_wmma_f32_16x16x4_f32: D = A x ones + C (row sums)
            d = __builtin_amdgcn_wmma_f32_16x16x4_f32(
                    false, a, false, ones, (short)0, d, false, false);
        }
        // lane 0 holds D rows 0-7 (col 0), lane 16 holds rows 8-15
        float s = d[0] + d[1] + d[2] + d[3] + d[4] + d[5] + d[6] + d[7];
        s += __shfl_down(s, 16, 32);
        if (tid == 0) wsB[(size_t)b * numTiles + tile] = s;
    }
}

// =====================================================================
// Kernel C: fixed-order final reduction + per-image normalization + mean.
// =====================================================================
DEV float blockRed256(float v, float* sm) {
    v = waveRed(v);
    const int lane = threadIdx.x & 31, wid = threadIdx.x >> 5;
    __syncthreads();                      // protect sm reuse across calls
    if (lane == 0) sm[wid] = v;
    __syncthreads();
    if (wid == 0) {
        float x = (lane < 8) ? sm[lane] : 0.f;
        x = waveRed(x);
        if (lane == 0) sm[0] = x;
    }
    __syncthreads();
    return sm[0];
}

__global__ void __launch_bounds__(256)
finalize_kernel(const float* __restrict__ wsB, const float* __restrict__ wsA,
                float* __restrict__ out, int B, int numTiles, int nBlkA)
{
    __shared__ float sm[8];
    const int tid = threadIdx.x;
    float cm = 0.f, rm = 0.f;
    for (int b = 0; b < B; ++b) {
        float sNeg = 0.f;
        for (int i = tid; i < numTiles; i += 256) sNeg += wsB[(size_t)b * numTiles + i];
        float sNp = 0.f, sCo = 0.f, sRs = 0.f;
        for (int i = tid; i < nBlkA; i += 256) {
            const float* p = wsA + ((size_t)b * nBlkA + i) * 3;
            sNp += p[0]; sCo += p[1]; sRs += p[2];
        }
        sNeg = blockRed256(sNeg, sm);
        sNp  = blockRed256(sNp,  sm);
        sCo  = blockRed256(sCo,  sm);
        sRs  = blockRed256(sRs,  sm);
        if (tid == 0) {
            const float clsSum = sNeg + sCo;
            const float cl = clsSum / fmaxf(sNp, 1.0f);
            const float rl = (sNp > 0.f) ? sRs / fmaxf(sNp * 4.0f, 1.0f) : 0.0f;
            cm += cl; rm += rl;
        }
    }
    if (tid == 0) { out[0] = cm / (float)B; out[1] = rm / (float)B; }
}

// =====================================================================
extern "C" void kernel_launch(void* const* d_in, const int* in_sizes, int n_in,
                              void* d_out, int out_size, void* d_ws, size_t ws_size,
                              hipStream_t stream)
{
    const float* cls = (const float*)d_in[0];   // [B,A,80]
    const float* reg = (const float*)d_in[1];   // [B,A,4]
    const float* anc = (const float*)d_in[2];   // [1,A,4]
    const float* lab = (const float*)d_in[3];   // [B,M,5]

    const int A = in_sizes[2] / 4;
    const int B = (A > 0) ? in_sizes[1] / (A * 4) : 1;
    const int M = (B > 0) ? in_sizes[3] / (B * 5) : 32;

    const int nBlkA    = (A + 255) / 256;
    const int numTiles = (A + TILE_A - 1) / TILE_A;

    float* w     = (float*)d_ws;
    float* tmask = w;                                   // B*A floats
    float* wsB   = w + (size_t)B * A;                   // B*numTiles floats
    float* wsA   = wsB + (size_t)B * numTiles;          // B*nBlkA*3 floats

    assign_kernel<<<dim3(nBlkA, B), 256, 0, stream>>>(cls, reg, anc, lab,
                                                      tmask, wsA, A, M, nBlkA);
    focal_neg_kernel<<<dim3(numTiles, B), 256, 0, stream>>>(cls, tmask, wsB,
                                                            A, numTiles);
    finalize_kernel<<<dim3(1), 256, 0, stream>>>(wsB, wsA, (float*)d_out,
                                                 B, numTiles, nBlkA);
    (void)n_in; (void)out_size; (void)ws_size;
}